// ZCANormSVDPI_No_Shrink_37151467110670
// MI455X (gfx1250) — compile-verified
//
#include <hip/hip_runtime.h>

typedef __attribute__((ext_vector_type(2))) float v2f;
typedef __attribute__((ext_vector_type(4))) float v4f;
typedef __attribute__((ext_vector_type(8))) float v8f;
typedef __attribute__((ext_vector_type(4))) int   v4i;

#define BATCH 32
#define CH    64
#define MLEN  16384
#define EPSV  1e-5f
#define DTOL  0.1f

// ---- workspace layout (float offsets) ----
#define OFF_GRAM 0u          // 32*64*64
#define OFF_RSUM 131072u     // 32*64
#define OFF_COV  133120u     // 32*64*64
#define OFF_VS   264192u     // 32*64*64  vs[b][j][c] = j-th eigenvector
#define OFF_LS   395264u     // 32*64
#define OFF_MU   397312u     // 32*64
#define OFF_LAM  399360u     // 32*64
#define OFF_SH   401408u     // 32*64*64
#define OFF_SMU  532480u     // 32*64
#define OFF_INT  534528u     // ints: condflags[64], flagarr[64], jstop[1]

#if __has_builtin(__builtin_amdgcn_global_load_async_to_lds_b128) && \
    __has_builtin(__builtin_amdgcn_s_wait_asynccnt)
#define USE_ASYNC_LDS 1
typedef __attribute__((address_space(1))) v4i* gv4i_p;   // global int4*
typedef __attribute__((address_space(3))) v4i* lv4i_p;   // LDS int4*
#endif

__device__ inline v8f wmma4(v2f a, v2f b, v8f c) {
  // D = A(16x4,f32) * B(4x16,f32) + C(16x16,f32)
  return __builtin_amdgcn_wmma_f32_16x16x4_f32(
      false, a, false, b, (short)0, c, false, false);
}

// -------------------- init: zero accumulators --------------------
__global__ void k_init(float* ws) {
  unsigned i = blockIdx.x * blockDim.x + threadIdx.x;
  if (i < OFF_RSUM + BATCH * CH) ws[i] = 0.0f;        // gram + rowsum
  if (i < 129) ((int*)(ws + OFF_INT))[i] = 0;         // condflags+flagarr+jstop
}

// -------------------- Gram + row sums (WMMA, async double-buffered) -------
// grid (16 splits, 32 batches), block 512 = 16 waves
#define TP 132
__global__ __launch_bounds__(512) void k_gram(const float* __restrict__ x,
                                              float* __restrict__ gram,
                                              float* __restrict__ rowsum) {
  __shared__ float tile[2][CH * TP];   // double-buffered 64 x 128 (padded)
  int b = blockIdx.y, sp = blockIdx.x;
  int tid = threadIdx.x, lane = tid & 31, wave = tid >> 5;
  int ti = wave >> 2, tj = wave & 3;          // 16x16 gram tile (ti,tj)
  int n = lane & 15, hf = lane >> 4;
  const float* xb = x + (size_t)b * CH * MLEN;

  v8f acc = {0.f,0.f,0.f,0.f,0.f,0.f,0.f,0.f};
  float rsum = 0.0f;

  // stage one 64x128 chunk into tile[buf] (async to LDS when available)
  auto issue = [&](int c, int buf) {
    int kbase = sp * 1024 + c * 128;
    #pragma unroll
    for (int it = 0; it < 4; ++it) {
      int fi = tid + it * 512;                // float4 index 0..2047
      int row = fi >> 5, c4 = fi & 31;
      const float* g = xb + (size_t)row * MLEN + kbase + c4 * 4;
      float* l = &tile[buf][row * TP + c4 * 4];
#ifdef USE_ASYNC_LDS
      __builtin_amdgcn_global_load_async_to_lds_b128(
          (gv4i_p)(size_t)g,
          (lv4i_p)(__attribute__((address_space(3))) void*)l, 0, 0);
#else
      *(v4f*)l = *(const v4f*)g;
#endif
    }
  };

  issue(0, 0);
  for (int c = 0; c < 8; ++c) {
    int buf = c & 1;
#ifdef USE_ASYNC_LDS
    __builtin_amdgcn_s_wait_asynccnt(0);   // my chunk-c loads landed in LDS
#endif
    __syncthreads();                       // everyone's landed; prev buf free
    if (c + 1 < 8) issue(c + 1, buf ^ 1);  // overlap next stage with WMMA

    if (tid < CH) {
      float s = 0.f;
      for (int k = 0; k < 128; ++k) s += tile[buf][tid * TP + k];
      rsum += s;
    }
    for (int kk = 0; kk < 128; kk += 4) {
      int ca = kk + 2 * hf;
      v2f a  = *(const v2f*)&tile[buf][(ti * 16 + n) * TP + ca];
      v2f bb = *(const v2f*)&tile[buf][(tj * 16 + n) * TP + ca];
      acc = wmma4(a, bb, acc);
    }
    __syncthreads();                       // done reading tile[buf]
  }

  float* gb = gram + (size_t)b * CH * CH;
  int mbase = 16 * ti + 8 * hf, nn = 16 * tj + n;
  #pragma unroll
  for (int v = 0; v < 8; ++v) atomicAdd(&gb[(mbase + v) * CH + nn], acc[v]);
  if (tid < CH) atomicAdd(&rowsum[b * CH + tid], rsum);
}

// -------------------- cov + parallel Jacobi eig + sort --------------------
// 32 blocks x 64 threads
__global__ __launch_bounds__(64) void k_eig(const float* __restrict__ gram,
                                            const float* __restrict__ rowsum,
                                            float* __restrict__ cov_out,
                                            float* __restrict__ vs,
                                            float* __restrict__ Ls,
                                            float* __restrict__ mu_out,
                                            int* __restrict__ condflags) {
  __shared__ float A[CH][CH];
  __shared__ float V[CH][CH];
  __shared__ float cs[32], sn[32];
  __shared__ int   pp[32], qq[32];
  __shared__ float d[CH];
  __shared__ int   idx[CH];
  __shared__ float mu[CH];
  int b = blockIdx.x, t = threadIdx.x;

  mu[t] = rowsum[b * CH + t] * (1.0f / MLEN);
  __syncthreads();
  const float* gb = gram + (size_t)b * CH * CH;
  for (int r = 0; r < CH; ++r) {
    float c = gb[r * CH + t] * (1.0f / MLEN) - mu[r] * mu[t];
    if (r == t) c += EPSV;
    A[r][t] = c;
    V[r][t] = (r == t) ? 1.0f : 0.0f;
  }
  __syncthreads();
  for (int r = 0; r < CH; ++r) cov_out[(size_t)b * CH * CH + r * CH + t] = A[r][t];

  for (int sweep = 0; sweep < 14; ++sweep) {
    for (int r = 0; r < 63; ++r) {
      if (t < 32) {
        int k1 = t, k2 = 63 - t;
        int p = (k1 == 0) ? 0 : ((k1 - 1 + r) % 63) + 1;
        int q = (k2 == 0) ? 0 : ((k2 - 1 + r) % 63) + 1;
        if (p > q) { int tmp = p; p = q; q = tmp; }
        pp[t] = p; qq[t] = q;
        float apq = A[p][q], app = A[p][p], aqq = A[q][q];
        float cc = 1.0f, ss = 0.0f;
        if (fabsf(apq) > 1e-12f) {
          float th = (aqq - app) / (2.0f * apq);
          float tt = ((th >= 0.f) ? 1.f : -1.f) / (fabsf(th) + sqrtf(1.f + th * th));
          cc = rsqrtf(1.0f + tt * tt);
          ss = tt * cc;
        }
        cs[t] = cc; sn[t] = ss;
      }
      __syncthreads();
      {
        int pr = t & 31, h = t >> 5;
        int p = pp[pr], q = qq[pr];
        float cc = cs[pr], ss = sn[pr];
        for (int i = h * 32; i < h * 32 + 32; ++i) {
          float aip = A[i][p], aiq = A[i][q];
          A[i][p] = cc * aip - ss * aiq;
          A[i][q] = ss * aip + cc * aiq;
          float vip = V[i][p], viq = V[i][q];
          V[i][p] = cc * vip - ss * viq;
          V[i][q] = ss * vip + cc * viq;
        }
      }
      __syncthreads();
      {
        int pr = t & 31, h = t >> 5;
        int p = pp[pr], q = qq[pr];
        float cc = cs[pr], ss = sn[pr];
        for (int i = h * 32; i < h * 32 + 32; ++i) {
          float api = A[p][i], aqi = A[q][i];
          A[p][i] = cc * api - ss * aqi;
          A[q][i] = ss * api + cc * aqi;
        }
      }
      __syncthreads();
    }
  }

  d[t] = A[t][t]; idx[t] = t;
  __syncthreads();
  for (int ph = 0; ph < CH; ++ph) {      // odd-even sort, descending
    int i0 = (ph & 1) + 2 * t;
    if (t < 32 && i0 + 1 < CH) {
      if (d[i0] < d[i0 + 1]) {
        float td2 = d[i0]; d[i0] = d[i0 + 1]; d[i0 + 1] = td2;
        int ti2 = idx[i0]; idx[i0] = idx[i0 + 1]; idx[i0 + 1] = ti2;
      }
    }
    __syncthreads();
  }

  Ls[b * CH + t] = d[t];
  mu_out[b * CH + t] = mu[t];
  for (int j = 0; j < CH; ++j)
    vs[(size_t)b * CH * CH + j * CH + t] = V[t][idx[j]];

  float tot = 0.f, cum = 0.f;
  for (int k = 0; k < CH; ++k) tot += d[k];
  for (int k = 0; k <= t; ++k) cum += d[k];
  int cflag = (cum / tot >= 1.0f - EPSV) || (d[t] <= EPSV);
  if (cflag) atomicOr(&condflags[t], 1);
}

// -------------------- deflated Rayleigh scan --------------------
// 32 blocks x 64 threads
__global__ __launch_bounds__(64) void k_lam(const float* __restrict__ cov,
                                            const float* __restrict__ vs,
                                            const float* __restrict__ Ls,
                                            float* __restrict__ lam_out,
                                            int* __restrict__ flagarr) {
  __shared__ float Cc[CH][CH];
  __shared__ float v[CH];
  __shared__ float w[CH];
  int b = blockIdx.x, t = threadIdx.x;
  for (int r = 0; r < CH; ++r) Cc[r][t] = cov[(size_t)b * CH * CH + r * CH + t];
  __syncthreads();
  for (int j = 0; j < CH; ++j) {
    v[t] = vs[(size_t)b * CH * CH + j * CH + t];
    __syncthreads();
    float s = 0.f;
    for (int k = 0; k < CH; ++k) s += Cc[t][k] * v[k];
    w[t] = s;
    __syncthreads();
    if (t == 0) {
      float vw = 0.f, vv = 0.f;
      for (int k = 0; k < CH; ++k) { vw += v[k] * w[k]; vv += v[k] * v[k]; }
      float lam = vw / vv;
      lam_out[b * CH + j] = lam;
      float Lj = Ls[b * CH + j];
      int bad = (lam < 0.0f) || (fabsf(lam - Lj) / Lj > DTOL);
      if (bad) atomicOr(&flagarr[j], 1);
    }
    __syncthreads();
    for (int k = 0; k < CH; ++k) Cc[t][k] -= w[t] * v[k];   // cov_c -= (cov_c v) v^T
    __syncthreads();
  }
}

// -------------------- global stop index --------------------
__global__ void k_stop(const int* __restrict__ condflags,
                       const int* __restrict__ flagarr,
                       int* __restrict__ jstop) {
  if (threadIdx.x == 0 && blockIdx.x == 0) {
    int counter = CH;
    for (int j = 0; j < CH; ++j) if (condflags[j]) { counter = j; break; }
    int js = counter;
    for (int j = 0; j < counter; ++j) if (flagarr[j]) { js = j; break; }
    *jstop = js;
  }
}

// -------------------- build S_hat and S_hat*mu --------------------
// 32 blocks x 256 threads
__global__ __launch_bounds__(256) void k_shat(const float* __restrict__ vs,
                                              const float* __restrict__ lam,
                                              const float* __restrict__ mu,
                                              const int* __restrict__ jstop_p,
                                              float* __restrict__ S,
                                              float* __restrict__ Smu) {
  __shared__ float Vsh[CH][CH];   // Vsh[j][c]
  __shared__ float coef[CH];
  __shared__ float mus[CH];
  __shared__ float tj[CH];
  int b = blockIdx.x, t = threadIdx.x;
  int js = *jstop_p;
  for (int i = t; i < CH * CH; i += 256) Vsh[i >> 6][i & 63] = vs[(size_t)b * CH * CH + i];
  if (t < CH) {
    float l = lam[b * CH + t];
    coef[t] = (l > 0.0f) ? rsqrtf(l) : 1.0f;   // lam_safe semantics
    mus[t] = mu[b * CH + t];
  }
  __syncthreads();
  for (int e = t; e < CH * CH; e += 256) {
    int r = e >> 6, c = e & 63;
    float s = 0.f;
    for (int j = 0; j < js; ++j) s += coef[j] * Vsh[j][r] * Vsh[j][c];
    S[(size_t)b * CH * CH + e] = s;
  }
  if (t < CH) {
    float s = 0.f;
    for (int c = 0; c < CH; ++c) s += Vsh[t][c] * mus[c];
    tj[t] = s;
  }
  __syncthreads();
  if (t < CH) {
    float s = 0.f;
    for (int j = 0; j < js; ++j) s += coef[j] * Vsh[j][t] * tj[j];
    Smu[b * CH + t] = s;
  }
}

// -------------------- Z = S_hat * x - S_hat*mu (WMMA) --------------------
// grid (128, 32), block 256 = 8 waves; each wave: 64x16 output strip
#define SP 68
__global__ __launch_bounds__(256) void k_out(const float* __restrict__ x,
                                             const float* __restrict__ S,
                                             const float* __restrict__ Smu,
                                             float* __restrict__ out) {
  __shared__ float Ssh[CH * SP];
  __shared__ float smu[CH];
  int b = blockIdx.y, nb = blockIdx.x;
  int t = threadIdx.x, lane = t & 31, wv = t >> 5;
  for (int i = t; i < CH * CH; i += 256)
    Ssh[(i >> 6) * SP + (i & 63)] = S[(size_t)b * CH * CH + i];
  if (t < CH) smu[t] = Smu[b * CH + t];
  __syncthreads();

  const float* xb = x + (size_t)b * CH * MLEN;
  float* ob = out + (size_t)b * CH * MLEN;
  int col0 = nb * 128 + wv * 16;
  int n = lane & 15, hf = lane >> 4;

  v8f acc0 = {0.f,0.f,0.f,0.f,0.f,0.f,0.f,0.f};
  v8f acc1 = acc0, acc2 = acc0, acc3 = acc0;

  #pragma unroll 4
  for (int k0 = 0; k0 < CH; k0 += 4) {
    int kr = k0 + 2 * hf;
    v2f bf;
    bf.x = xb[(size_t)kr * MLEN + col0 + n];
    bf.y = xb[(size_t)(kr + 1) * MLEN + col0 + n];
    int ca = k0 + 2 * hf;
    v2f a0 = *(const v2f*)&Ssh[(0  + n) * SP + ca];
    v2f a1 = *(const v2f*)&Ssh[(16 + n) * SP + ca];
    v2f a2 = *(const v2f*)&Ssh[(32 + n) * SP + ca];
    v2f a3 = *(const v2f*)&Ssh[(48 + n) * SP + ca];
    acc0 = wmma4(a0, bf, acc0);
    acc1 = wmma4(a1, bf, acc1);
    acc2 = wmma4(a2, bf, acc2);
    acc3 = wmma4(a3, bf, acc3);
  }

  #pragma unroll
  for (int v = 0; v < 8; ++v) {
    int m = v + 8 * hf;
    ob[(size_t)(m)      * MLEN + col0 + n] = acc0[v] - smu[m];
    ob[(size_t)(16 + m) * MLEN + col0 + n] = acc1[v] - smu[16 + m];
    ob[(size_t)(32 + m) * MLEN + col0 + n] = acc2[v] - smu[32 + m];
    ob[(size_t)(48 + m) * MLEN + col0 + n] = acc3[v] - smu[48 + m];
  }
}

extern "C" void kernel_launch(void* const* d_in, const int* in_sizes, int n_in,
                              void* d_out, int out_size, void* d_ws, size_t ws_size,
                              hipStream_t stream) {
  const float* x = (const float*)d_in[0];
  float* out = (float*)d_out;
  float* ws = (float*)d_ws;

  float* gram   = ws + OFF_GRAM;
  float* rowsum = ws + OFF_RSUM;
  float* cov    = ws + OFF_COV;
  float* vs     = ws + OFF_VS;
  float* Ls     = ws + OFF_LS;
  float* mu     = ws + OFF_MU;
  float* lam    = ws + OFF_LAM;
  float* Sh     = ws + OFF_SH;
  float* Smu    = ws + OFF_SMU;
  int*   iflags = (int*)(ws + OFF_INT);
  int*   condflags = iflags;
  int*   flagarr   = iflags + 64;
  int*   jstop     = iflags + 128;

  k_init<<<520, 256, 0, stream>>>(ws);
  k_gram<<<dim3(16, 32), 512, 0, stream>>>(x, gram, rowsum);
  k_eig<<<32, 64, 0, stream>>>(gram, rowsum, cov, vs, Ls, mu, condflags);
  k_lam<<<32, 64, 0, stream>>>(cov, vs, Ls, lam, flagarr);
  k_stop<<<1, 64, 0, stream>>>(condflags, flagarr, jstop);
  k_shat<<<32, 256, 0, stream>>>(vs, lam, mu, jstop, Sh, Smu);
  k_out<<<dim3(128, 32), 256, 0, stream>>>(x, Sh, Smu, out);
}